// SparseBPLinear_360777253007
// MI455X (gfx1250) — compile-verified
//
#include <hip/hip_runtime.h>

typedef __attribute__((ext_vector_type(2))) float v2f;
typedef __attribute__((ext_vector_type(8))) float v8f;

#define P_DIM 1024
#define Q_DIM 1024
#define KB    8
#define BS    64
#define NCHN  8192      // Q*KB (in-channels == k extent)
#define QC    32        // q-blocks per chunk
#define KC    256       // k per chunk = QC*KB
#define NCHUNK (NCHN / KC)

__global__ __launch_bounds__(256)
void sbpl_wmma_kernel(const float* __restrict__ x,
                      const float* __restrict__ u,
                      const float* __restrict__ s,
                      const float* __restrict__ v,
                      float* __restrict__ out)
{
    // Staging for B^T chunk: wt[k_local][n], k_local in [0,256), n in [0,16). 16 KB.
    __shared__ float wt[KC * 16];

    const int pp   = blockIdx.x;        // 0..511 : 16-column output stripe
    const int tid  = threadIdx.x;       // 256 threads = 8 wave32
    const int lane = tid & 31;
    const int wave = tid >> 5;
    const int m    = wave & 3;          // M tile (rows 16m..16m+15 of x/out)
    const int h    = wave >> 2;         // split-K half (0/1)
    const int l15  = lane & 15;
    const int kh   = lane >> 4;         // 0/1 : K sub-pair selector (WMMA f32 A/B layout)

    // ---- assembly-phase mapping: thread -> (p_sel, q-in-chunk, i-pair) ----
    const int blk  = tid >> 2;          // 0..63 : (qq, psel)
    const int qq   = blk >> 1;          // 0..31
    const int psel = blk & 1;           // which of the two p's
    const int i0   = (tid & 3) * 2;     // i-pair {i0, i0+1}
    const int p    = pp * 2 + psel;

    const float* xrow = x + (size_t)(m * 16 + l15) * NCHN;

    v8f acc = (v8f)0.0f;

    for (int ch = 0; ch < NCHUNK; ++ch) {
        const int qbase = ch * QC;

        // -------- assemble Wt[k][n] = (U diag(s) V)[i(n), l] for this chunk --------
        {
            const int q = qbase + qq;
            const size_t base = (size_t)p * Q_DIM + q;
            const float4* U0 = (const float4*)(u + (base * KB + i0) * KB);
            const float4* U1 = (const float4*)(u + (base * KB + i0 + 1) * KB);
            const float4* S4 = (const float4*)(s + base * KB);
            const float4* V4 = (const float4*)(v + base * KB * KB);

            // prefetch next chunk's U/V for this thread (global_prefetch_b8)
            if (ch + 1 < NCHUNK) {
                const size_t nb = (size_t)p * Q_DIM + (q + QC);
                __builtin_prefetch(u + (nb * KB + i0) * KB, 0, 0);
                __builtin_prefetch(v + nb * KB * KB, 0, 0);
                __builtin_prefetch(v + nb * KB * KB + 32, 0, 0);
            }

            float uu0[8], uu1[8], ss[8];
            float4 t;
            t = U0[0]; uu0[0]=t.x; uu0[1]=t.y; uu0[2]=t.z; uu0[3]=t.w;
            t = U0[1]; uu0[4]=t.x; uu0[5]=t.y; uu0[6]=t.z; uu0[7]=t.w;
            t = U1[0]; uu1[0]=t.x; uu1[1]=t.y; uu1[2]=t.z; uu1[3]=t.w;
            t = U1[1]; uu1[4]=t.x; uu1[5]=t.y; uu1[6]=t.z; uu1[7]=t.w;
            t = S4[0]; ss[0]=t.x;  ss[1]=t.y;  ss[2]=t.z;  ss[3]=t.w;
            t = S4[1]; ss[4]=t.x;  ss[5]=t.y;  ss[6]=t.z;  ss[7]=t.w;

            float a0[8], a1[8];
            #pragma unroll
            for (int l = 0; l < 8; ++l) { a0[l] = 0.f; a1[l] = 0.f; }

            #pragma unroll
            for (int j = 0; j < 8; ++j) {
                const float w0 = uu0[j] * ss[j];
                const float w1 = uu1[j] * ss[j];
                float vv[8];
                t = V4[2*j];   vv[0]=t.x; vv[1]=t.y; vv[2]=t.z; vv[3]=t.w;
                t = V4[2*j+1]; vv[4]=t.x; vv[5]=t.y; vv[6]=t.z; vv[7]=t.w;
                #pragma unroll
                for (int l = 0; l < 8; ++l) {
                    a0[l] = __builtin_fmaf(w0, vv[l], a0[l]);
                    a1[l] = __builtin_fmaf(w1, vv[l], a1[l]);
                }
            }

            // Wt row = qq*8 + l (local k), col n = psel*8 + i ; i-pair is contiguous
            const int ncol = psel * 8 + i0;   // even -> 8B-aligned float2 store
            #pragma unroll
            for (int l = 0; l < 8; ++l) {
                float2 pr; pr.x = a0[l]; pr.y = a1[l];
                *(float2*)&wt[(qq * 8 + l) * 16 + ncol] = pr;
            }
        }
        __syncthreads();

        // -------- WMMA: acc += X[16m.., k] * Wt[k, n], k over this wave's half --------
        const int kglob0 = ch * KC + h * (KC / 2);   // global k base for this wave
        const int kloc0  = h * (KC / 2);             // LDS k base
        #pragma unroll 4
        for (int kk = 0; kk < KC / 2; kk += 4) {
            // A (16x4 f32): lane -> row l15, K pair {2*kh, 2*kh+1} of this step
            v2f a = *(const v2f*)(xrow + kglob0 + kk + 2 * kh);
            // B (4x16 f32): lane -> col l15, K rows {2*kh, 2*kh+1}
            const int kl = kloc0 + kk + 2 * kh;
            v2f b;
            b.x = wt[kl * 16 + l15];
            b.y = wt[(kl + 1) * 16 + l15];
            acc = __builtin_amdgcn_wmma_f32_16x16x4_f32(
                false, a, false, b, (short)0, acc, false, false);
        }
        __syncthreads();
    }

    // -------- split-K reduction (wave halves) through LDS, then store --------
    float* red = wt;   // reuse staging LDS (4 KB needed)
    if (h == 1) {
        #pragma unroll
        for (int r = 0; r < 8; ++r)
            red[(m * 8 + r) * 32 + lane] = acc[r];
    }
    __syncthreads();
    if (h == 0) {
        #pragma unroll
        for (int r = 0; r < 8; ++r)
            acc[r] += red[(m * 8 + r) * 32 + lane];
        const int col = pp * 16 + l15;
        #pragma unroll
        for (int r = 0; r < 8; ++r) {
            const int row = m * 16 + r + 8 * kh;   // C/D layout: lanes16-31 -> M+8
            out[(size_t)row * NCHN + col] = acc[r];
        }
    }
}

extern "C" void kernel_launch(void* const* d_in, const int* in_sizes, int n_in,
                              void* d_out, int out_size, void* d_ws, size_t ws_size,
                              hipStream_t stream) {
    const float* x = (const float*)d_in[0];
    const float* u = (const float*)d_in[1];
    const float* s = (const float*)d_in[2];
    const float* v = (const float*)d_in[3];
    float* out = (float*)d_out;
    (void)in_sizes; (void)n_in; (void)out_size; (void)d_ws; (void)ws_size;

    dim3 grid(P_DIM / 2);   // 512 workgroups, one 16-column output stripe each
    dim3 block(256);        // 8 wave32
    sbpl_wmma_kernel<<<grid, block, 0, stream>>>(x, u, s, v, out);
}